// NeighborhoodAttentionModule_6923487281775
// MI455X (gfx1250) — compile-verified
//
#include <hip/hip_runtime.h>

#define HIDDEN 256
#define ATTN_D 64
#define KNBR   64
#define TILE_M 16
#define CPAD   260   // padded LDS row stride (floats)

typedef float v2f __attribute__((ext_vector_type(2)));
typedef float v8f __attribute__((ext_vector_type(8)));

// ---------------------------------------------------------------------------
// Tile GEMM inner loop: acc(16x16) += A(16 x kdim, LDS, stride lda) *
// W(kdim x N global row-major, col n). Uses V_WMMA_F32_16X16X4_F32.
// A layout per ISA: lanes 0-15 hold K pair {0,1}, lanes 16-31 hold {2,3}.
// ---------------------------------------------------------------------------
__device__ __forceinline__ v8f wmma_gemm_acc(const float* A, int lda,
                                             const float* __restrict__ W, int ldb,
                                             int kdim, int n, int lane, v8f acc) {
  const int m  = lane & 15;
  const int kh = (lane >> 4) << 1;  // 0 or 2
  #pragma unroll 4
  for (int kk = 0; kk < kdim; kk += 4) {
    const int k0 = kk + kh;
    v2f a, b;
    a.x = A[m * lda + k0];
    a.y = A[m * lda + k0 + 1];
    b.x = W[(size_t)k0 * ldb + n];
    b.y = W[(size_t)(k0 + 1) * ldb + n];
    acc = __builtin_amdgcn_wmma_f32_16x16x4_f32(false, a, false, b,
                                                (short)0, acc, false, false);
  }
  return acc;
}

// ---------------------------------------------------------------------------
// Kernel 1: M2[e][d] = sum_a Wq[e][a] * Wk[d][a]   (256x256, tiny)
// ---------------------------------------------------------------------------
__global__ __launch_bounds__(256)
void build_m2(const float* __restrict__ Wq, const float* __restrict__ Wk,
              float* __restrict__ M2) {
  const int e = blockIdx.x, d = threadIdx.x;
  const float4* a = (const float4*)(Wq + e * ATTN_D);
  const float4* b = (const float4*)(Wk + d * ATTN_D);
  float acc = 0.f;
  #pragma unroll
  for (int i = 0; i < ATTN_D / 4; ++i) {
    float4 av = a[i], bv = b[i];
    acc += av.x * bv.x + av.y * bv.y + av.z * bv.z + av.w * bv.w;
  }
  M2[e * HIDDEN + d] = acc;
}

// ---------------------------------------------------------------------------
// Kernel 2: fused attention + gate + output projection + layernorm.
// One block = 16 rows of B. 256 threads = 8 wave32.
// ---------------------------------------------------------------------------
__global__ __launch_bounds__(256)
void fused_kernel(const float* __restrict__ C,   // [B,256] center_emb
                  const float* __restrict__ E,   // [B,64,256] neighbor_embs
                  const float* __restrict__ Wt,  // [B,64] neighbor_weights
                  const float* __restrict__ M2,  // [256,256] Wq@Wk^T
                  const float* __restrict__ Wg, const float* __restrict__ bg,
                  const float* __restrict__ Wo, const float* __restrict__ bo,
                  const float* __restrict__ gamma, const float* __restrict__ beta,
                  float* __restrict__ out) {
  __shared__ float Cst[TILE_M * CPAD];    // center tile (A-operand, padded)
  __shared__ float QK [TILE_M * HIDDEN];  // qk tile; reused for enriched tile
  __shared__ float CTX[TILE_M * CPAD];    // context tile (A-operand, padded)
  __shared__ float EBUF[KNBR * CPAD];     // staged neighbor rows for one b
  __shared__ float SC[KNBR];
  __shared__ float PW[KNBR];

  const int tid  = threadIdx.x;
  const int lane = tid & 31;
  const int wave = tid >> 5;
  const int row0 = blockIdx.x * TILE_M;
  const int nl   = lane & 15;
  const int n0   = wave * 16 + nl;
  const int n1   = (wave + 8) * 16 + nl;
  const int rb   = (lane >> 4) << 3;      // accumulator row base (0 or 8)

  // ---- load center tile into LDS -----------------------------------------
  {
    const int r  = tid >> 4;
    const int c0 = (tid & 15) * 16;
    const float4* src = (const float4*)(C + (size_t)(row0 + r) * HIDDEN + c0);
    #pragma unroll
    for (int i = 0; i < 4; ++i) {
      float4 v = src[i];
      float* d = &Cst[r * CPAD + c0 + i * 4];
      d[0] = v.x; d[1] = v.y; d[2] = v.z; d[3] = v.w;
    }
  }
  __syncthreads();

  // ---- GEMM1: QK = Ctile @ M2 (WMMA) -------------------------------------
  {
    v8f a0 = {}, a1 = {};
    a0 = wmma_gemm_acc(Cst, CPAD, M2, HIDDEN, HIDDEN, n0, lane, a0);
    a1 = wmma_gemm_acc(Cst, CPAD, M2, HIDDEN, HIDDEN, n1, lane, a1);
    #pragma unroll
    for (int i = 0; i < 8; ++i) {
      QK[(rb + i) * HIDDEN + n0] = a0[i];
      QK[(rb + i) * HIDDEN + n1] = a1[i];
    }
  }
  __syncthreads();

  // ---- attention: single HBM pass over E, one b at a time ----------------
  for (int b = 0; b < TILE_M; ++b) {
    const size_t ebase = (size_t)(row0 + b) * (KNBR * HIDDEN);
    // stage E[b] (64x256) into LDS, fully coalesced float4 loads
    #pragma unroll
    for (int i = 0; i < 16; ++i) {
      const int idx = tid + 256 * i;        // float4 index 0..4095
      const int j   = idx >> 6;             // 64 float4 per row
      const int c   = (idx & 63) * 4;
      float4 v = *(const float4*)(E + ebase + (size_t)j * HIDDEN + c);
      float* d = &EBUF[j * CPAD + c];
      d[0] = v.x; d[1] = v.y; d[2] = v.z; d[3] = v.w;
    }
    __syncthreads();
    // scores: 4 threads per neighbor, each dots 64 dims, shuffle-reduce
    {
      const int j = tid >> 2;
      const int q = tid & 3;
      const float* er = &EBUF[j * CPAD + q * 64];
      const float* qk = &QK[b * HIDDEN + q * 64];
      float acc = 0.f;
      #pragma unroll 8
      for (int i = 0; i < 64; ++i) acc += er[i] * qk[i];
      acc += __shfl_xor(acc, 1);
      acc += __shfl_xor(acc, 2);
      if (q == 0) {
        float w = Wt[(size_t)(row0 + b) * KNBR + j];
        SC[j] = acc * 0.125f + __logf(fmaxf(w, 1e-6f));  // /sqrt(64)+log conf
      }
    }
    __syncthreads();
    // softmax probabilities (64 threads, redundant max/sum is cheap)
    if (tid < KNBR) {
      float mx = -1e30f;
      for (int j = 0; j < KNBR; ++j) mx = fmaxf(mx, SC[j]);
      float z = 0.f;
      for (int j = 0; j < KNBR; ++j) z += __expf(SC[j] - mx);
      PW[tid] = __expf(SC[tid] - mx) / z;
    }
    __syncthreads();
    // context_raw[b][d] = sum_j p_j * E[b][j][d]
    {
      float acc = 0.f;
      for (int j = 0; j < KNBR; ++j) acc += PW[j] * EBUF[j * CPAD + tid];
      CTX[b * CPAD + tid] = acc;
    }
    __syncthreads();
  }

  // ---- GEMM2: gate = sigmoid(Ctile@Wg + bg); CTX *= gate (WMMA) ----------
  {
    v8f a0 = {}, a1 = {};
    a0 = wmma_gemm_acc(Cst, CPAD, Wg, HIDDEN, HIDDEN, n0, lane, a0);
    a1 = wmma_gemm_acc(Cst, CPAD, Wg, HIDDEN, HIDDEN, n1, lane, a1);
    const float b0 = bg[n0], b1 = bg[n1];
    #pragma unroll
    for (int i = 0; i < 8; ++i) {
      float g0 = 1.f / (1.f + __expf(-(a0[i] + b0)));
      float g1 = 1.f / (1.f + __expf(-(a1[i] + b1)));
      CTX[(rb + i) * CPAD + n0] *= g0;
      CTX[(rb + i) * CPAD + n1] *= g1;
    }
  }
  __syncthreads();

  // ---- GEMM3: enriched = [Ctile|CTX]@Wo + bo + Ctile (WMMA, K=512) -------
  {
    v8f a0 = {}, a1 = {};
    a0 = wmma_gemm_acc(Cst, CPAD, Wo, HIDDEN, HIDDEN, n0, lane, a0);
    a1 = wmma_gemm_acc(Cst, CPAD, Wo, HIDDEN, HIDDEN, n1, lane, a1);
    const float* Wo2 = Wo + (size_t)HIDDEN * HIDDEN;   // rows 256..511
    a0 = wmma_gemm_acc(CTX, CPAD, Wo2, HIDDEN, HIDDEN, n0, lane, a0);
    a1 = wmma_gemm_acc(CTX, CPAD, Wo2, HIDDEN, HIDDEN, n1, lane, a1);
    const float b0 = bo[n0], b1 = bo[n1];
    #pragma unroll
    for (int i = 0; i < 8; ++i) {
      const int r = rb + i;
      QK[r * HIDDEN + n0] = a0[i] + b0 + Cst[r * CPAD + n0];
      QK[r * HIDDEN + n1] = a1[i] + b1 + Cst[r * CPAD + n1];
    }
  }
  __syncthreads();

  // ---- LayerNorm: 16 lanes per row, shuffle reductions -------------------
  {
    const int g = tid >> 4;
    const int l = tid & 15;
    float x[16];
    const float* er = &QK[g * HIDDEN + l * 16];
    float s = 0.f;
    #pragma unroll
    for (int i = 0; i < 16; ++i) { x[i] = er[i]; s += x[i]; }
    s += __shfl_xor(s, 1); s += __shfl_xor(s, 2);
    s += __shfl_xor(s, 4); s += __shfl_xor(s, 8);
    const float mu = s * (1.f / HIDDEN);
    float v = 0.f;
    #pragma unroll
    for (int i = 0; i < 16; ++i) { float d = x[i] - mu; v += d * d; }
    v += __shfl_xor(v, 1); v += __shfl_xor(v, 2);
    v += __shfl_xor(v, 4); v += __shfl_xor(v, 8);
    const float rs = rsqrtf(v * (1.f / HIDDEN) + 1e-5f);
    float* o = out + (size_t)(row0 + g) * HIDDEN + l * 16;
    #pragma unroll
    for (int i = 0; i < 16; ++i) {
      const int d = l * 16 + i;
      o[i] = (x[i] - mu) * rs * gamma[d] + beta[d];
    }
  }
}

// ---------------------------------------------------------------------------
extern "C" void kernel_launch(void* const* d_in, const int* in_sizes, int n_in,
                              void* d_out, int out_size, void* d_ws, size_t ws_size,
                              hipStream_t stream) {
  const float* C     = (const float*)d_in[0];
  const float* E     = (const float*)d_in[1];
  const float* Wt    = (const float*)d_in[2];
  const float* Wq    = (const float*)d_in[3];
  const float* Wk    = (const float*)d_in[4];
  const float* Wg    = (const float*)d_in[5];
  const float* bg    = (const float*)d_in[6];
  const float* Wo    = (const float*)d_in[7];
  const float* bo    = (const float*)d_in[8];
  const float* gamma = (const float*)d_in[9];
  const float* beta  = (const float*)d_in[10];
  float* out = (float*)d_out;
  float* M2  = (float*)d_ws;               // 256*256*4 = 256 KB scratch

  const int B = in_sizes[0] / HIDDEN;      // 16384
  build_m2<<<HIDDEN, HIDDEN, 0, stream>>>(Wq, Wk, M2);
  fused_kernel<<<B / TILE_M, 256, 0, stream>>>(C, E, Wt, M2, Wg, bg, Wo, bo,
                                               gamma, beta, out);
}